// SignedConv_57286273794081
// MI455X (gfx1250) — compile-verified
//
#include <hip/hip_runtime.h>
#include <cstdint>
#include <cstddef>

typedef __attribute__((ext_vector_type(2))) float v2f;
typedef __attribute__((ext_vector_type(8))) float v8f;

#define N_NODES 100000
#define F 64

// ---------------------------------------------------------------------------
// Edge scatter: 64 consecutive threads per edge, one f32 feature each.
// sum[dst][f] += x[src][f]; cnt[dst] += 1 (lane f==0).
// Atomics are contiguous per edge -> good L2 channel locality; x rows are
// 256B coalesced reads that stay hot in the 192MB L2.
// ---------------------------------------------------------------------------
__global__ void __launch_bounds__(256)
edge_scatter_kernel(const long long* __restrict__ ei,   // [2*E] int64: src then dst
                    const float* __restrict__ x,        // [N, 64]
                    float* __restrict__ sum,            // [N, 64]
                    float* __restrict__ cnt,            // [N]
                    long long E) {
    long long idx = (long long)blockIdx.x * blockDim.x + threadIdx.x;
    long long e = idx >> 6;
    int f = (int)(idx & 63);
    if (e >= E) return;
    long long src = ei[e];
    long long dst = ei[E + e];
    float v = x[src * F + f];
    unsafeAtomicAdd(&sum[dst * F + f], v);          // global_atomic_add_f32 (no return)
    if (f == 0) unsafeAtomicAdd(&cnt[dst], 1.0f);
}

// ---------------------------------------------------------------------------
// Fused mean + 4 GEMMs + bias + concat using V_WMMA_F32_16X16X4_F32.
// One wave handles a 16-node tile x 128 output columns (8 C tiles).
//   A (16x4 f32): lane 0-15 row M=lane, V0=K, V1=K+1; lanes 16-31 K+2,K+3.
//   B (4x16 f32): lane%16 = column N (=output j), same K split by lane>=16.
//   C/D (16x16 f32): lane%16 = N, VGPR v = M=v (lanes<16) / M=8+v (lanes>=16).
// ---------------------------------------------------------------------------
__global__ void __launch_bounds__(256)
signed_conv_gemm_kernel(const float* __restrict__ x,
                        const float* __restrict__ sum_pos,
                        const float* __restrict__ cnt_pos,
                        const float* __restrict__ sum_neg,
                        const float* __restrict__ cnt_neg,
                        const float* __restrict__ W_pos,
                        const float* __restrict__ W_pos_cc,
                        const float* __restrict__ b_pos_cc,
                        const float* __restrict__ W_neg,
                        const float* __restrict__ W_neg_cc,
                        const float* __restrict__ b_neg_cc,
                        float* __restrict__ out) {
    const int lane = threadIdx.x & 31;
    const int wave = threadIdx.x >> 5;
    const int tileBase = (blockIdx.x * 8 + wave) * 16;   // first node row of tile
    const int m  = lane & 15;      // A row within tile / B,C,D column within tile
    const int hi = lane >> 4;      // K-half select for A/B
    const int kSel = hi * 2;

    // Clamp row for loads (keeps EXEC all-ones for WMMA); stores are guarded.
    int r = tileBase + m;
    int rc = (r < N_NODES) ? r : (N_NODES - 1);

    const float rp = 1.0f / fmaxf(cnt_pos[rc], 1.0f);
    const float rn = 1.0f / fmaxf(cnt_neg[rc], 1.0f);

    v8f accP[4], accN[4];
#pragma unroll
    for (int t = 0; t < 4; ++t) {
        const int j = t * 16 + m;
        const float bp = b_pos_cc[j];
        const float bn = b_neg_cc[j];
#pragma unroll
        for (int i = 0; i < 8; ++i) { accP[t][i] = bp; accN[t][i] = bn; }
    }

    const float* xrow  = x       + (size_t)rc * F;
    const float* sprow = sum_pos + (size_t)rc * F;
    const float* snrow = sum_neg + (size_t)rc * F;

#pragma unroll 4
    for (int k = 0; k < F; k += 4) {
        const v2f ax = *(const v2f*)(xrow  + k + kSel);
        const v2f ap = *(const v2f*)(sprow + k + kSel) * rp;   // fold mean into A
        const v2f an = *(const v2f*)(snrow + k + kSel) * rn;
#pragma unroll
        for (int t = 0; t < 4; ++t) {
            const int j = t * 16 + m;
            const v2f bwp  = *(const v2f*)(W_pos    + j * F + k + kSel);
            const v2f bwpc = *(const v2f*)(W_pos_cc + j * F + k + kSel);
            const v2f bwn  = *(const v2f*)(W_neg    + j * F + k + kSel);
            const v2f bwnc = *(const v2f*)(W_neg_cc + j * F + k + kSel);
            accP[t] = __builtin_amdgcn_wmma_f32_16x16x4_f32(
                false, ap, false, bwp,  (short)0, accP[t], false, false);
            accP[t] = __builtin_amdgcn_wmma_f32_16x16x4_f32(
                false, ax, false, bwpc, (short)0, accP[t], false, false);
            accN[t] = __builtin_amdgcn_wmma_f32_16x16x4_f32(
                false, an, false, bwn,  (short)0, accN[t], false, false);
            accN[t] = __builtin_amdgcn_wmma_f32_16x16x4_f32(
                false, ax, false, bwnc, (short)0, accN[t], false, false);
        }
    }

    // Store: out[row, 0:64] = pos, out[row, 64:128] = neg.
#pragma unroll
    for (int t = 0; t < 4; ++t) {
        const int j = t * 16 + m;
#pragma unroll
        for (int v = 0; v < 8; ++v) {
            const int row = tileBase + v + 8 * hi;
            if (row < N_NODES) {
                out[(size_t)row * 128 + j]      = accP[t][v];
                out[(size_t)row * 128 + 64 + j] = accN[t][v];
            }
        }
    }
}

// ---------------------------------------------------------------------------
// Launch
// ---------------------------------------------------------------------------
extern "C" void kernel_launch(void* const* d_in, const int* in_sizes, int n_in,
                              void* d_out, int out_size, void* d_ws, size_t ws_size,
                              hipStream_t stream) {
    const float*     x        = (const float*)d_in[0];
    const long long* pos_ei   = (const long long*)d_in[1];  // int64 [2, E]
    const long long* neg_ei   = (const long long*)d_in[2];  // int64 [2, E]
    const float*     W_pos    = (const float*)d_in[3];
    const float*     W_pos_cc = (const float*)d_in[4];
    const float*     b_pos_cc = (const float*)d_in[5];
    const float*     W_neg    = (const float*)d_in[6];
    const float*     W_neg_cc = (const float*)d_in[7];
    const float*     b_neg_cc = (const float*)d_in[8];
    float* out = (float*)d_out;

    const long long E = (long long)in_sizes[1] / 2;

    // Workspace layout: sum_pos[N*64] | sum_neg[N*64] | cnt_pos[N] | cnt_neg[N]
    float* sum_pos = (float*)d_ws;
    float* sum_neg = sum_pos + (size_t)N_NODES * F;
    float* cnt_pos = sum_neg + (size_t)N_NODES * F;
    float* cnt_neg = cnt_pos + N_NODES;

    const size_t zero_bytes =
        ((size_t)2 * N_NODES * F + (size_t)2 * N_NODES) * sizeof(float);
    hipMemsetAsync(d_ws, 0, zero_bytes, stream);

    const long long work = E * 64;
    const int sblocks = (int)((work + 255) / 256);
    edge_scatter_kernel<<<sblocks, 256, 0, stream>>>(pos_ei, x, sum_pos, cnt_pos, E);
    edge_scatter_kernel<<<sblocks, 256, 0, stream>>>(neg_ei, x, sum_neg, cnt_neg, E);

    const int tiles   = (N_NODES + 15) / 16;
    const int gblocks = (tiles + 7) / 8;     // 8 waves per block, 16 nodes per wave
    signed_conv_gemm_kernel<<<gblocks, 256, 0, stream>>>(
        x, sum_pos, cnt_pos, sum_neg, cnt_neg,
        W_pos, W_pos_cc, b_pos_cc, W_neg, W_neg_cc, b_neg_cc, out);
}